// GCNNodeRegression_76390288327438
// MI455X (gfx1250) — compile-verified
//
#include <hip/hip_runtime.h>
#include <hip/hip_bf16.h>

#define NN 50000
#define NE 800000
#define DD 512

typedef float v2f __attribute__((ext_vector_type(2)));
typedef float v8f __attribute__((ext_vector_type(8)));

// ---- gfx1250 async global->LDS copy path (ASYNCcnt-tracked), guarded so the
// ---- file still compiles to the synchronous path on toolchains without it.
#if defined(__has_builtin)
#  if __has_builtin(__builtin_amdgcn_global_load_async_to_lds_b128) && \
      __has_builtin(__builtin_amdgcn_s_wait_asynccnt)
#    define HAVE_ASYNC_LDS 1
#  endif
#endif
#ifndef HAVE_ASYNC_LDS
#  define HAVE_ASYNC_LDS 0
#endif

// exact pointee type from the toolchain's builtin signature:
//   'int __attribute__((__vector_size__(4*sizeof(int)))) __device__ *'  (AS1)
typedef int v4i_b128 __attribute__((vector_size(16)));
typedef __attribute__((address_space(1))) v4i_b128 g_v4i_t;   // global
typedef __attribute__((address_space(3))) v4i_b128 l_v4i_t;   // LDS

__device__ __forceinline__ void copy16_to_lds(float* lds_dst, const float* gsrc) {
#if HAVE_ASYNC_LDS
    __builtin_amdgcn_global_load_async_to_lds_b128((g_v4i_t*)gsrc,
                                                   (l_v4i_t*)lds_dst, 0, 0);
#else
    *(float4*)lds_dst = *(const float4*)gsrc;
#endif
}

__device__ __forceinline__ void wait_lds_copies() {
#if HAVE_ASYNC_LDS
    __builtin_amdgcn_s_wait_asynccnt(0);
#endif
}

// ---------------------------------------------------------------- utilities

__global__ void zero_f32(float* __restrict__ p, long n) {
    long i = (long)blockIdx.x * blockDim.x + threadIdx.x;
    long stride = (long)gridDim.x * blockDim.x;
    for (; i < n; i += stride) p[i] = 0.0f;
}

__global__ void degree_kernel(const int* __restrict__ src, const int* __restrict__ dst,
                              float* __restrict__ deg_out, float* __restrict__ deg_in, int e) {
    int i = blockIdx.x * blockDim.x + threadIdx.x;
    if (i < e) {
        atomicAdd(&deg_out[src[i]], 1.0f);
        atomicAdd(&deg_in[dst[i]], 1.0f);
    }
}

// in-place: deg -> rsqrt(max(deg,1))
__global__ void norm_kernel(float* __restrict__ p, int n) {
    int i = blockIdx.x * blockDim.x + threadIdx.x;
    if (i < n) p[i] = rsqrtf(fmaxf(p[i], 1.0f));
}

// x[i][c] = relu(agg[i][c]*norm_in[i] + b[c])   (one block per node row)
__global__ __launch_bounds__(128)
void act_kernel(const float* __restrict__ agg, const float* __restrict__ norm_in,
                const float* __restrict__ bias, float* __restrict__ out) {
    const int node = blockIdx.x;
    const int c = threadIdx.x * 4;
    const float s = norm_in[node];
    float4 v = *(const float4*)(agg + (size_t)node * DD + c);
    float4 b = *(const float4*)(bias + c);
    v.x = fmaxf(fmaf(v.x, s, b.x), 0.0f);
    v.y = fmaxf(fmaf(v.y, s, b.y), 0.0f);
    v.z = fmaxf(fmaf(v.z, s, b.z), 0.0f);
    v.w = fmaxf(fmaf(v.w, s, b.w), 0.0f);
    *(float4*)(out + (size_t)node * DD + c) = v;
}

// ---------------------------------------------------------------- tiled WMMA GEMM
// out[m][n] = ( A[m][:] @ W[:][n] ) * postNorm[m]
// A: [M x 512] row-major, W: [512 x 512] row-major, out: [M x 512].
// Block: 128 threads = 4 waves (2x2), block tile 64x64.
// K staged 16-wide via double-buffered LDS filled with async global->LDS DMA:
// stage(next) overlaps with the 16 f32 WMMAs on the current buffer.

#define BM 64
#define BN 64
#define BK 16

__global__ __launch_bounds__(128)
void gemm_gcn(const float* __restrict__ A, const float* __restrict__ W,
              float* __restrict__ out, const float* __restrict__ postNorm, int M) {
    __shared__ float As[2][BM][20];   // padded stride -> conflict-free frag reads
    __shared__ float Bs[2][BK][BN];

    const int tid  = threadIdx.x;
    const int lane = tid & 31;
    const int wave = tid >> 5;
    const int wm = (wave >> 1) * 32;      // wave's M offset in block tile
    const int wn = (wave & 1) * 32;       // wave's N offset in block tile
    const int ml = lane & 15;
    const int khalf = (lane >> 4) * 2;    // lanes 0-15 -> K+0/1, lanes 16-31 -> K+2/3

    const int m0 = blockIdx.y * BM;
    const int n0 = blockIdx.x * BN;

    // per-thread staging coordinates (2 x b128 for A, 2 x b128 for B)
    int ar[2], ac[2], agr[2], br[2], bc[2];
    #pragma unroll
    for (int i = 0; i < 2; ++i) {
        int f = tid + i * 128;            // 0..255
        ar[i] = f >> 2;                   // A row 0..63
        ac[i] = (f & 3) << 2;             // A col 0,4,8,12
        int g = m0 + ar[i]; if (g >= M) g = M - 1;   // clamp (stores guarded)
        agr[i] = g;
        br[i] = f >> 4;                   // B row 0..15
        bc[i] = (f & 15) << 2;            // B col 0..60
    }

    v8f acc00 = {}, acc01 = {}, acc10 = {}, acc11 = {};

    // issue stage 0
    #pragma unroll
    for (int i = 0; i < 2; ++i) {
        copy16_to_lds(&As[0][ar[i]][ac[i]], A + (size_t)agr[i] * DD + ac[i]);
        copy16_to_lds(&Bs[0][br[i]][bc[i]], W + (size_t)br[i] * DD + n0 + bc[i]);
    }

    int buf = 0;
    for (int k0 = 0; k0 < DD; k0 += BK) {
        wait_lds_copies();                // my DMA into `buf` done
        __syncthreads();                  // everyone's DMA done + prev compute done

        if (k0 + BK < DD) {               // kick off next stage into the other buffer
            const int kn = k0 + BK;
            #pragma unroll
            for (int i = 0; i < 2; ++i) {
                copy16_to_lds(&As[buf ^ 1][ar[i]][ac[i]],
                              A + (size_t)agr[i] * DD + kn + ac[i]);
                copy16_to_lds(&Bs[buf ^ 1][br[i]][bc[i]],
                              W + (size_t)(kn + br[i]) * DD + n0 + bc[i]);
            }
        }

        // ---- 4 K-steps of 4: full-precision f32 WMMA (16x16x4), 16 per stage
        #pragma unroll
        for (int kk = 0; kk < BK; kk += 4) {
            const int kq = kk + khalf;
            v2f a0, a1, b0, b1;
            a0[0] = As[buf][wm + ml][kq];          a0[1] = As[buf][wm + ml][kq + 1];
            a1[0] = As[buf][wm + 16 + ml][kq];     a1[1] = As[buf][wm + 16 + ml][kq + 1];
            b0[0] = Bs[buf][kq][wn + ml];          b0[1] = Bs[buf][kq + 1][wn + ml];
            b1[0] = Bs[buf][kq][wn + 16 + ml];     b1[1] = Bs[buf][kq + 1][wn + 16 + ml];
            acc00 = __builtin_amdgcn_wmma_f32_16x16x4_f32(false, a0, false, b0, (short)0, acc00, false, false);
            acc01 = __builtin_amdgcn_wmma_f32_16x16x4_f32(false, a0, false, b1, (short)0, acc01, false, false);
            acc10 = __builtin_amdgcn_wmma_f32_16x16x4_f32(false, a1, false, b0, (short)0, acc10, false, false);
            acc11 = __builtin_amdgcn_wmma_f32_16x16x4_f32(false, a1, false, b1, (short)0, acc11, false, false);
        }
        buf ^= 1;
        __syncthreads();                  // all waves done reading before next overwrite
    }

    // ---- epilogue: C/D layout: VGPR r -> M = r + (lane<16 ? 0 : 8), N = lane&15
    const int rowHalf = (lane >> 4) * 8;
    #pragma unroll
    for (int mi = 0; mi < 2; ++mi) {
        #pragma unroll
        for (int ni = 0; ni < 2; ++ni) {
            const v8f acc = mi == 0 ? (ni == 0 ? acc00 : acc01)
                                    : (ni == 0 ? acc10 : acc11);
            const int col = n0 + wn + ni * 16 + ml;
            #pragma unroll
            for (int r = 0; r < 8; ++r) {
                int row = m0 + wm + mi * 16 + rowHalf + r;
                if (row < M)
                    out[(size_t)row * DD + col] = acc[r] * postNorm[row];
            }
        }
    }
}

// ---------------------------------------------------------------- edge scatter-add
// agg[dst[e]][:] += h[src[e]][:]   (atomics land in 192MB L2; agg buffer is 102MB)
// 128 threads per edge, float4 gather + 4 global_atomic_add_f32 each.
__global__ __launch_bounds__(256)
void edge_scatter(const float* __restrict__ h, const int* __restrict__ src,
                  const int* __restrict__ dst, float* __restrict__ agg, int e) {
    int eid = blockIdx.x * 2 + (threadIdx.x >> 7);
    if (eid >= e) return;
    int c = (threadIdx.x & 127) << 2;
    const float4 v = *(const float4*)(h + (size_t)src[eid] * DD + c);
    float* d = agg + (size_t)dst[eid] * DD + c;
    atomicAdd(d + 0, v.x);
    atomicAdd(d + 1, v.y);
    atomicAdd(d + 2, v.z);
    atomicAdd(d + 3, v.w);
}

// ---------------------------------------------------------------- fused final layer
// out[i] = sum_c relu(agg[i][c]*norm_in[i] + b2[c]) * Wf[c] + bf
__global__ __launch_bounds__(256)
void final_dot(const float* __restrict__ agg, const float* __restrict__ norm_in,
               const float* __restrict__ b2, const float* __restrict__ Wf,
               const float* __restrict__ bf, float* __restrict__ out, int n) {
    const int lane = threadIdx.x & 31;
    const int node = blockIdx.x * (blockDim.x >> 5) + (threadIdx.x >> 5);
    if (node >= n) return;
    const float ni = norm_in[node];
    const float* row = agg + (size_t)node * DD;
    float sum = 0.0f;
    #pragma unroll
    for (int c0 = 0; c0 < DD; c0 += 128) {
        int c = c0 + lane * 4;
        float4 v = *(const float4*)(row + c);
        float4 b = *(const float4*)(b2 + c);
        float4 w = *(const float4*)(Wf + c);
        sum = fmaf(fmaxf(fmaf(v.x, ni, b.x), 0.0f), w.x, sum);
        sum = fmaf(fmaxf(fmaf(v.y, ni, b.y), 0.0f), w.y, sum);
        sum = fmaf(fmaxf(fmaf(v.z, ni, b.z), 0.0f), w.z, sum);
        sum = fmaf(fmaxf(fmaf(v.w, ni, b.w), 0.0f), w.w, sum);
    }
    #pragma unroll
    for (int off = 16; off > 0; off >>= 1)
        sum += __shfl_down(sum, off, 32);
    if (lane == 0) out[node] = sum + bf[0];
}

// ---------------------------------------------------------------- launch

extern "C" void kernel_launch(void* const* d_in, const int* in_sizes, int n_in,
                              void* d_out, int out_size, void* d_ws, size_t ws_size,
                              hipStream_t stream) {
    const float* features = (const float*)d_in[0];
    const int*   src      = (const int*)  d_in[1];
    const int*   dst      = (const int*)  d_in[2];
    const float* W1       = (const float*)d_in[3];
    const float* b1       = (const float*)d_in[4];
    const float* W2       = (const float*)d_in[5];
    const float* b2       = (const float*)d_in[6];
    const float* Wf       = (const float*)d_in[7];
    const float* bf       = (const float*)d_in[8];
    float* out = (float*)d_out;

    // workspace layout (floats): [norm_out NN][norm_in NN][bufA NN*DD][bufB NN*DD]
    float* norm_out = (float*)d_ws;
    float* norm_in  = norm_out + NN;
    float* bufA     = norm_out + 2 * NN;          // 16B aligned (2*NN*4 = 400000)
    float* bufB     = bufA + (size_t)NN * DD;

    const long nd = (long)NN * DD;
    dim3 gemm_grid(DD / BN, (NN + BM - 1) / BM);  // 8 x 782

    // degrees -> norms (norm_out & norm_in computed in one shot on the 2*NN range)
    zero_f32<<<2048, 256, 0, stream>>>(norm_out, 2L * NN);
    degree_kernel<<<(NE + 255) / 256, 256, 0, stream>>>(src, dst, norm_out, norm_in, NE);
    norm_kernel<<<(2 * NN + 255) / 256, 256, 0, stream>>>(norm_out, 2 * NN);

    // layer 1: bufA = (features @ W1) * norm_out ; bufB = scatter-add over edges
    gemm_gcn<<<gemm_grid, 128, 0, stream>>>(features, W1, bufA, norm_out, NN);
    zero_f32<<<2048, 256, 0, stream>>>(bufB, nd);
    edge_scatter<<<(NE + 1) / 2, 256, 0, stream>>>(bufA, src, dst, bufB, NE);

    // layer 2: bufA = relu(bufB*norm_in + b1); bufB = (bufA @ W2) * norm_out
    act_kernel<<<NN, 128, 0, stream>>>(bufB, norm_in, b1, bufA);
    gemm_gcn<<<gemm_grid, 128, 0, stream>>>(bufA, W2, bufB, norm_out, NN);
    zero_f32<<<2048, 256, 0, stream>>>(bufA, nd);
    edge_scatter<<<(NE + 1) / 2, 256, 0, stream>>>(bufB, src, dst, bufA, NE);

    // fused relu(bufA*norm_in + b2) @ Wf + bf
    final_dot<<<(NN + 7) / 8, 256, 0, stream>>>(bufA, norm_in, b2, Wf, bf, out, NN);
}